// UserSpotConv_26585847562449
// MI455X (gfx1250) — compile-verified
//
#include <hip/hip_runtime.h>
#include <stdint.h>

// ---------------------------------------------------------------------------
// LightGCN-style bipartite aggregation for MI455X (gfx1250, wave32).
// Memory-bound scatter/gather: no WMMA applicable (dense GEMM would be ~300x
// more FLOPs than the roofline-optimal sparse path). Optimizations target the
// 23.3 TB/s HBM / 192 MB L2 hierarchy:
//   * edge stream read once with non-temporal loads (keep tables L2-resident)
//   * one wave32 per 32-edge chunk; per-edge indices broadcast via ds_bpermute
//   * full 512B feature rows moved as coalesced b128 loads (1 float4/lane)
//   * global_prefetch_b8 covering the whole 32KB chunk working set
//   * native fp32 global atomics (L2 atomic units) for the scatter-add
// ---------------------------------------------------------------------------

typedef int v4i __attribute__((ext_vector_type(4)));

__device__ __forceinline__ void atomic_fadd_agent(float* p, float v) {
#if __has_builtin(__builtin_amdgcn_global_atomic_fadd_f32)
    __builtin_amdgcn_global_atomic_fadd_f32(
        (__attribute__((address_space(1))) float*)p, v);
#else
#warning "ATOMIC-PROBE: __builtin_amdgcn_global_atomic_fadd_f32 unavailable, using atomicAdd fallback"
    atomicAdd(p, v);
#endif
}

// ---------------------------------------------------------------------------
// Main scatter kernel: one wave processes chunks of 32 edges.
// ---------------------------------------------------------------------------
__global__ __launch_bounds__(256) void usc_scatter_kernel(
    const float* __restrict__ spot_x,   // [M, 128]
    const float* __restrict__ user_x,   // [N, 128]
    const int*   __restrict__ u_idx,    // [E]
    const int*   __restrict__ s_idx,    // [E]
    const int*   __restrict__ udeg,     // [N]
    const int*   __restrict__ sdeg,     // [M]
    float*       __restrict__ spot_out, // [M, 128]
    float*       __restrict__ user_out, // [N, 128]
    int E)
{
    const int lane  = threadIdx.x & 31;
    const int gwave = (int)((blockIdx.x * blockDim.x + threadIdx.x) >> 5);
    const int nwave = (int)((gridDim.x * blockDim.x) >> 5);
    const int foff  = lane << 2;   // this lane's float4 slot within a 128-row

    for (int base = gwave * 32; base < E; base += nwave * 32) {
        const int cnt = min(32, E - base);
        // Clamp so inactive-tail lanes load a valid (unused) edge.
        const int e = base + min(lane, cnt - 1);

        // Edge stream: read once, non-temporal (don't pollute L2).
        const int u = __builtin_nontemporal_load(u_idx + e);
        const int s = __builtin_nontemporal_load(s_idx + e);

        // Symmetric GCN normalization (degrees are small ints; fp32 exact).
        const float inv = 1.0f / sqrtf((float)(udeg[u] * sdeg[s]));

        // Prefetch this lane's edge's feature rows in full (512B = 4 lines
        // each) -> global_prefetch_b8. Across the wave this covers all 32
        // rows of both matrices (2 x 16KB) before the broadcast loop.
        const float* srow = spot_x + (size_t)s * 128;
        const float* urow = user_x + (size_t)u * 128;
#pragma unroll
        for (int q = 0; q < 4; ++q) {
            __builtin_prefetch(srow + q * 32, 0, 3);
            __builtin_prefetch(urow + q * 32, 0, 3);
        }

        for (int j = 0; j < cnt; ++j) {
            const int   uj = __shfl(u,   j, 32);
            const int   sj = __shfl(s,   j, 32);
            const float w  = __shfl(inv, j, 32);

            const float4 sv = *(const float4*)(spot_x + sj * 128 + foff);
            const float4 uv = *(const float4*)(user_x + uj * 128 + foff);

            float* uo = user_out + uj * 128 + foff;
            float* so = spot_out + sj * 128 + foff;

            atomic_fadd_agent(uo + 0, sv.x * w);
            atomic_fadd_agent(uo + 1, sv.y * w);
            atomic_fadd_agent(uo + 2, sv.z * w);
            atomic_fadd_agent(uo + 3, sv.w * w);

            atomic_fadd_agent(so + 0, uv.x * w);
            atomic_fadd_agent(so + 1, uv.y * w);
            atomic_fadd_agent(so + 2, uv.z * w);
            atomic_fadd_agent(so + 3, uv.w * w);
        }
    }
}

// ---------------------------------------------------------------------------
// Degree histogram: b128 non-temporal index loads (4 edges/iter), int atomics.
// ---------------------------------------------------------------------------
__global__ __launch_bounds__(256) void usc_degree_kernel(
    const int* __restrict__ u_idx,
    const int* __restrict__ s_idx,
    int* __restrict__ udeg,
    int* __restrict__ sdeg,
    int E)
{
    const int i      = (int)(blockIdx.x * blockDim.x + threadIdx.x);
    const int stride = (int)(gridDim.x * blockDim.x);
    const int E4     = E >> 2;

    for (int q = i; q < E4; q += stride) {
        const v4i uu = __builtin_nontemporal_load((const v4i*)u_idx + q);
        const v4i ss = __builtin_nontemporal_load((const v4i*)s_idx + q);
        atomicAdd(&udeg[uu.x], 1);
        atomicAdd(&udeg[uu.y], 1);
        atomicAdd(&udeg[uu.z], 1);
        atomicAdd(&udeg[uu.w], 1);
        atomicAdd(&sdeg[ss.x], 1);
        atomicAdd(&sdeg[ss.y], 1);
        atomicAdd(&sdeg[ss.z], 1);
        atomicAdd(&sdeg[ss.w], 1);
    }
    // Tail (E not a multiple of 4).
    for (int e = (E4 << 2) + i; e < E; e += stride) {
        atomicAdd(&udeg[__builtin_nontemporal_load(u_idx + e)], 1);
        atomicAdd(&sdeg[__builtin_nontemporal_load(s_idx + e)], 1);
    }
}

// ---------------------------------------------------------------------------
// Zero fill (b128 stores).
// ---------------------------------------------------------------------------
__global__ __launch_bounds__(256) void usc_zero_kernel(uint32_t* __restrict__ p,
                                                       long n)
{
    const long i      = (long)blockIdx.x * blockDim.x + threadIdx.x;
    const long stride = (long)gridDim.x * blockDim.x;
    for (long k = i * 4; k < n; k += stride * 4) {
        if (k + 4 <= n) {
            *(uint4*)(p + k) = make_uint4(0u, 0u, 0u, 0u);
        } else {
            for (long t = k; t < n; ++t) p[t] = 0u;
        }
    }
}

extern "C" void kernel_launch(void* const* d_in, const int* in_sizes, int n_in,
                              void* d_out, int out_size, void* d_ws, size_t ws_size,
                              hipStream_t stream)
{
    const float* spot_x = (const float*)d_in[0];   // [M,128] f32
    const float* user_x = (const float*)d_in[1];   // [N,128] f32
    const int*   edges  = (const int*)d_in[2];     // [2,E] i32

    const int M = in_sizes[0] / 128;
    const int N = in_sizes[1] / 128;
    const int E = in_sizes[2] / 2;

    const int* u_idx = edges;        // row 0
    const int* s_idx = edges + E;    // row 1

    float* spot_out = (float*)d_out;                    // [M,128] first
    float* user_out = spot_out + (size_t)M * 128;       // [N,128] second

    // Workspace: degree arrays (contiguous): udeg[N] then sdeg[M].
    int* udeg = (int*)d_ws;
    int* sdeg = udeg + N;

    // 1) Zero degrees ((N+M)*4 bytes, well under ws_size) and outputs.
    {
        const long n = (long)N + (long)M;
        const int blocks = (int)((n + 1023) / 1024);
        usc_zero_kernel<<<blocks, 256, 0, stream>>>((uint32_t*)udeg, n);
    }
    {
        const long n = (long)out_size;
        const int blocks = (int)((n + 1023) / 1024);
        usc_zero_kernel<<<blocks, 256, 0, stream>>>((uint32_t*)d_out, n);
    }

    // 2) Degree histogram.
    usc_degree_kernel<<<1024, 256, 0, stream>>>(u_idx, s_idx, udeg, sdeg, E);

    // 3) Normalized bidirectional scatter-add.
    usc_scatter_kernel<<<2048, 256, 0, stream>>>(
        spot_x, user_x, u_idx, s_idx, udeg, sdeg, spot_out, user_out, E);
}